// SidewaysTime_1245540516184
// MI455X (gfx1250) — compile-verified
//
#include <hip/hip_runtime.h>

// ---------------------------------------------------------------------------
// Problem constants (from the reference): B=32, N = 128*16 = 2048 features.
// out[b] = [ flat (2048) | tril(outer(flat,flat)) (N(N+1)/2 = 2098176) ]
// Store-bandwidth bound: ~268.8 MB of fp32 output, 1 mul per 4 bytes.
// Strategy: one wave32 per 16x16 outer-product tile via V_WMMA_F32_16X16X4_F32,
// non-temporal coalesced stores into the packed triangular layout.
// ---------------------------------------------------------------------------

typedef __attribute__((ext_vector_type(2))) float v2f;
typedef __attribute__((ext_vector_type(4))) float v4f;
typedef __attribute__((ext_vector_type(8))) float v8f;

#define NFEAT   2048
#define TRI     ((size_t)NFEAT * (NFEAT + 1) / 2)   // 2,098,176
#define OUT_PER ((size_t)NFEAT + TRI)               // 2,100,224 floats per sample
#define NTILES  8256                                // 128*129/2 triangular 16x16 tiles
#define BATCHSZ 32

// --- linear part: copy flat features (256 KB total) ------------------------
__global__ void copy_flat_kernel(const float* __restrict__ in,
                                 float* __restrict__ out) {
    // 32 samples * 512 float4 each = 16384 threads
    unsigned idx = blockIdx.x * blockDim.x + threadIdx.x;
    unsigned b   = idx >> 9;            // 512 float4 per sample
    unsigned k   = (idx & 511u) << 2;   // float offset
    v4f v = *(const v4f*)(in + (size_t)b * NFEAT + k);
    float* dst = out + (size_t)b * OUT_PER + k;
    __builtin_nontemporal_store(v.x, dst + 0);
    __builtin_nontemporal_store(v.y, dst + 1);
    __builtin_nontemporal_store(v.z, dst + 2);
    __builtin_nontemporal_store(v.w, dst + 3);
}

// --- triangular outer-product tiles ----------------------------------------
// blockIdx.y = sample, one wave per tile. All waves full => EXEC all-1s at WMMA.
__global__ void outer_tri_kernel(const float* __restrict__ in,
                                 float* __restrict__ out) {
    const unsigned lane = threadIdx.x & 31u;
    const unsigned b    = blockIdx.y;
    const unsigned t    = blockIdx.x * (blockDim.x >> 5) + (threadIdx.x >> 5);

    // Decode triangular tile id t -> (ti, tj), ti >= tj, row-major lower tri.
    unsigned ti = (unsigned)((sqrtf(8.0f * (float)t + 1.0f) - 1.0f) * 0.5f);
    while ((ti + 1u) * (ti + 2u) / 2u <= t) ++ti;   // fp round-off fixups
    while (ti * (ti + 1u) / 2u > t)        --ti;
    const unsigned tj = t - ti * (ti + 1u) / 2u;

    // Load the 16-float x (row) and y (col) segments; input is L2-resident.
    const float* flat = in + (size_t)b * NFEAT;
    const unsigned l15 = lane & 15u;
    const float xv = flat[ti * 16u + l15];
    const float yv = flat[tj * 16u + l15];
    const bool lo = lane < 16u;

    // A(16x4): only column k=0 nonzero -> A VGPR0 lanes 0..15 = x[i].
    // B(4x16): only row    k=0 nonzero -> B VGPR0 lanes 0..15 = y[j].
    // D[i][j] = x[i]*y[j]  (rank-1 update via the matrix pipe).
    v2f av; av.x = lo ? xv : 0.0f; av.y = 0.0f;
    v2f bv; bv.x = lo ? yv : 0.0f; bv.y = 0.0f;
    v8f c = {};
    v8f d = __builtin_amdgcn_wmma_f32_16x16x4_f32(
        /*neg_a=*/false, av, /*neg_b=*/false, bv,
        /*c_mod=*/(short)0, c, /*reuse_a=*/false, /*reuse_b=*/false);

    // D layout: VGPR v -> row (v) for lanes 0-15, row (v+8) for lanes 16-31;
    // column = lane & 15. Scatter into packed triangle: N + I*(I+1)/2 + J.
    float* outNL = out + (size_t)b * OUT_PER + NFEAT;
    const unsigned Jg    = tj * 16u + l15;
    const unsigned Mbase = ti * 16u + (lo ? 0u : 8u);

    if (ti != tj) {
        // Full off-diagonal tile: 8 stores/lane, each VGPR = two 64B row runs.
#pragma unroll
        for (int v = 0; v < 8; ++v) {
            const unsigned I   = Mbase + (unsigned)v;
            const unsigned off = I * (I + 1u) / 2u + Jg;
            __builtin_nontemporal_store(d[v], outNL + off);
        }
    } else {
        // Diagonal tile: keep only I >= J (predication is AFTER the WMMA).
#pragma unroll
        for (int v = 0; v < 8; ++v) {
            const unsigned I = Mbase + (unsigned)v;
            if (I >= Jg) {
                const unsigned off = I * (I + 1u) / 2u + Jg;
                __builtin_nontemporal_store(d[v], outNL + off);
            }
        }
    }
}

extern "C" void kernel_launch(void* const* d_in, const int* in_sizes, int n_in,
                              void* d_out, int out_size, void* d_ws, size_t ws_size,
                              hipStream_t stream) {
    const float* data = (const float*)d_in[0];   // (32,128,16) fp32, contiguous
    float* out = (float*)d_out;                  // 32 * 2,100,224 fp32

    // Linear copy: 16384 threads = 64 blocks.
    copy_flat_kernel<<<64, 256, 0, stream>>>(data, out);

    // 8256 tiles/sample, 8 waves (tiles) per 256-thread block -> 1032 blocks,
    // gridDim.y = 32 samples. Every wave full => EXEC all-1s for WMMA.
    dim3 grid(NTILES / 8, BATCHSZ, 1);
    outer_tri_kernel<<<grid, 256, 0, stream>>>(data, out);
}